// CTRGCNEncoder_89335319757569
// MI455X (gfx1250) — compile-verified
//
#include <hip/hip_runtime.h>
#include <cmath>

// ---------------------------------------------------------------------------
// CTR-GCN encoder for MI455X (gfx1250), fp32 throughout using
// V_WMMA_F32_16X16X4_F32 for all channel-mixing GEMMs.
// Guard-free inner loops: block0's cin=3 is zero-padded to 4 once up front,
// TCN/residual weights are BN-prescaled and repacked for contiguous b64
// fragment loads.
//
// Input flattening assumption (insertion "dict order", recursive):
//   0:x 1:adj 2..18:block0{key_w,key_b,query_w,query_b,static_w,static_b,
//   gcn_bn_g,gcn_bn_b,alpha,tcn_w,tcn_b,tcn_bn_g,tcn_bn_b,res_w,res_b,
//   res_bn_g,res_bn_b} 19..35:block1{...} 36:proj_w 37:proj_b
// ---------------------------------------------------------------------------

typedef float v2f __attribute__((ext_vector_type(2)));
typedef float v8f __attribute__((ext_vector_type(8)));

#define DEVI __device__ __forceinline__

constexpr int BB   = 64;
constexpr int TT   = 512;
constexpr int JJ   = 17;
constexpr int NP   = TT * JJ;        // 8704 positions (t,j), 68*128
constexpr int PSTR = (TT + 2) * JJ;  // 8738 padded positions (temporal halo)

// fp32 WMMA: D(16x16) = A(16x4) x B(4x16) + C
DEVI v8f wmma4(v2f a, v2f b, v8f c) {
  return __builtin_amdgcn_wmma_f32_16x16x4_f32(false, a, false, b, (short)0, c,
                                               false, false);
}

DEVI v2f load2(const float* p) {  // 8B-aligned contiguous pair -> b64 load
  return *reinterpret_cast<const v2f*>(p);
}

DEVI void store8(float* dst, v8f v, float add, bool relu) {
  float4 lo4, hi4;
  float t0 = v[0] + add, t1 = v[1] + add, t2 = v[2] + add, t3 = v[3] + add;
  float t4 = v[4] + add, t5 = v[5] + add, t6 = v[6] + add, t7 = v[7] + add;
  if (relu) {
    t0 = fmaxf(t0, 0.f); t1 = fmaxf(t1, 0.f); t2 = fmaxf(t2, 0.f);
    t3 = fmaxf(t3, 0.f); t4 = fmaxf(t4, 0.f); t5 = fmaxf(t5, 0.f);
    t6 = fmaxf(t6, 0.f); t7 = fmaxf(t7, 0.f);
  }
  lo4 = make_float4(t0, t1, t2, t3);
  hi4 = make_float4(t4, t5, t6, t7);
  *reinterpret_cast<float4*>(dst)     = lo4;
  *reinterpret_cast<float4*>(dst + 4) = hi4;
}

// ---------------------------------------------------------------------------
// Packing kernels (run once per call, tiny)
// ---------------------------------------------------------------------------

// x (B,3,NP) -> xpad (B,4,NP), channel 3 = 0
__global__ void pack_x_kernel(const float* __restrict__ x, float* __restrict__ xp) {
  const int idx = blockIdx.x * blockDim.x + threadIdx.x;
  const int total = BB * 4 * NP;
  if (idx >= total) return;
  const int n = idx % NP;
  const int c = (idx / NP) & 3;
  const int b = idx / (4 * NP);
  xp[idx] = (c < 3) ? x[((size_t)b * 3 + c) * NP + n] : 0.f;
}

// (cout,3) weight -> (cout,4), optional per-o scale
__global__ void pack_w3_kernel(const float* __restrict__ w, float* __restrict__ o4,
                               const float* __restrict__ scale, int use_scale,
                               int cout) {
  const int idx = blockIdx.x * blockDim.x + threadIdx.x;
  if (idx >= cout * 4) return;
  const int c = idx & 3, o = idx >> 2;
  float v = (c < 3) ? w[o * 3 + c] : 0.f;
  if (use_scale) v *= scale[o];
  o4[idx] = v;
}

// tcn_w (cout,cout,3) -> twp[dt][o][c] scaled by tg[o]
__global__ void pack_tw_kernel(const float* __restrict__ tw, const float* __restrict__ tg,
                               float* __restrict__ twp, int cout) {
  const int idx = blockIdx.x * blockDim.x + threadIdx.x;
  const int total = 3 * cout * cout;
  if (idx >= total) return;
  const int c  = idx % cout;
  const int o  = (idx / cout) % cout;
  const int dt = idx / (cout * cout);
  twp[idx] = tg[o] * tw[((size_t)o * cout + c) * 3 + dt];
}

// res_w (cout,cin) -> rwp scaled by rg[o]  (cin already multiple of 4)
__global__ void pack_rw_kernel(const float* __restrict__ rw, const float* __restrict__ rg,
                               float* __restrict__ rwp, int cin, int cout) {
  const int idx = blockIdx.x * blockDim.x + threadIdx.x;
  if (idx >= cin * cout) return;
  rwp[idx] = rg[idx / cin] * rw[idx];
}

// ---------------------------------------------------------------------------
// Mean over T: xt[b,c,j] = mean_t h[b,c,t,j].  One block per (b,c).
// ---------------------------------------------------------------------------
__global__ void meanT_kernel(const float* __restrict__ H, float* __restrict__ xt) {
  const int bc  = blockIdx.x;
  const int tid = threadIdx.x;          // 0..271
  const int tg  = tid / JJ;             // 0..15
  const int j   = tid % JJ;
  __shared__ float s[272];
  const float* base = H + (size_t)bc * NP;
  float acc = 0.f;
  for (int t = tg; t < TT; t += 16) acc += base[t * JJ + j];
  s[tid] = acc;
  __syncthreads();
  if (tid < JJ) {
    float tot = 0.f;
    #pragma unroll
    for (int g = 0; g < 16; ++g) tot += s[g * JJ + tid];
    xt[(size_t)bc * JJ + tid] = tot * (1.0f / (float)TT);
  }
}

// ---------------------------------------------------------------------------
// key/query 1x1 conv on the T-mean
// ---------------------------------------------------------------------------
__global__ void kq_kernel(const float* __restrict__ xt,
                          const float* __restrict__ KW, const float* __restrict__ KB,
                          const float* __restrict__ QW, const float* __restrict__ QB,
                          float* __restrict__ Ko, float* __restrict__ Qo,
                          int cin, int cout) {
  const int idx   = blockIdx.x * blockDim.x + threadIdx.x;
  const int total = BB * cout * JJ;
  if (idx >= total) return;
  const int j = idx % JJ;
  const int o = (idx / JJ) % cout;
  const int b = idx / (JJ * cout);
  float ak = KB[o], aq = QB[o];
  const float* xb = xt + (size_t)b * cin * JJ + j;
  for (int c = 0; c < cin; ++c) {
    const float v = xb[c * JJ];
    ak += KW[o * cin + c] * v;
    aq += QW[o * cin + c] * v;
  }
  Ko[idx] = ak;
  Qo[idx] = aq;
}

// ---------------------------------------------------------------------------
// Attention softmax -> dyn (B,17,17)
// ---------------------------------------------------------------------------
__global__ void attn_kernel(const float* __restrict__ Ko,
                            const float* __restrict__ Qo,
                            float* __restrict__ dyn, int cout) {
  const int b   = blockIdx.x;
  const int tid = threadIdx.x;          // 0..288
  const int j   = tid / JJ;
  const int kk  = tid % JJ;
  __shared__ float slog[289];
  __shared__ float sexp[289];
  float acc = 0.f;
  const float* qb = Qo + (size_t)b * cout * JJ;
  const float* kb = Ko + (size_t)b * cout * JJ;
  for (int c = 0; c < cout; ++c) acc += qb[c * JJ + j] * kb[c * JJ + kk];
  acc *= (1.0f / sqrtf((float)cout));
  slog[tid] = acc;
  __syncthreads();
  float mx = -3.4e38f;
  #pragma unroll
  for (int i = 0; i < JJ; ++i) mx = fmaxf(mx, slog[j * JJ + i]);
  const float e = __expf(acc - mx);
  sexp[tid] = e;
  __syncthreads();
  float sum = 0.f;
  #pragma unroll
  for (int i = 0; i < JJ; ++i) sum += sexp[j * JJ + i];
  dyn[(size_t)b * (JJ * JJ) + tid] = e / sum;
}

// ---------------------------------------------------------------------------
// P = static_w @ h  (per batch, guard-free).  Wave tile: 32 pos x 64 ch.
// A frag (16x4 f32): a.x = elem(m = lane&15, K = k0 + (lane>>4)*2), a.y K+1.
// B frag (4x16 f32): b.x = elem(K = k0 + (lane>>4)*2, n = lane&15), b.y K+1.
// D: vgpr r -> M = r + (lane>>4)*8, N = lane&15.
// ---------------------------------------------------------------------------
template <int CIN, int COUT>
__global__ __launch_bounds__(128) void gemm_p_kernel(const float* __restrict__ H,
                                                     const float* __restrict__ W,
                                                     float* __restrict__ P) {
  const int lane = threadIdx.x & 31;
  const int wv   = threadIdx.x >> 5;
  const int lo = lane & 15, hi = lane >> 4;
  const int m0 = (blockIdx.x * 4 + wv) * 32;   // 32-position wave tile
  const int o0 = blockIdx.y * 64;
  const int b  = blockIdx.z;
  const float* Hb = H + (size_t)b * CIN * NP;

  v8f acc0[4] = {v8f{}, v8f{}, v8f{}, v8f{}};
  v8f acc1[4] = {v8f{}, v8f{}, v8f{}, v8f{}};

  #pragma unroll 2
  for (int k0 = 0; k0 < CIN; k0 += 4) {
    const int kb = k0 + hi * 2;
    v2f a0, a1;
    a0.x = Hb[(size_t)kb * NP + m0 + lo];
    a0.y = Hb[(size_t)(kb + 1) * NP + m0 + lo];
    a1.x = Hb[(size_t)kb * NP + m0 + 16 + lo];
    a1.y = Hb[(size_t)(kb + 1) * NP + m0 + 16 + lo];
    v2f bw[4];
    #pragma unroll
    for (int s = 0; s < 4; ++s)
      bw[s] = load2(W + (o0 + s * 16 + lo) * CIN + kb);   // contiguous b64
    #pragma unroll
    for (int s = 0; s < 4; ++s) acc0[s] = wmma4(a0, bw[s], acc0[s]);
    #pragma unroll
    for (int s = 0; s < 4; ++s) acc1[s] = wmma4(a1, bw[s], acc1[s]);
  }
  #pragma unroll
  for (int s = 0; s < 4; ++s) {
    const int o = o0 + s * 16 + lo;
    float* dst = P + ((size_t)b * COUT + o) * NP + m0 + hi * 8;
    store8(dst, acc0[s], 0.f, false);
    store8(dst + 16, acc1[s], 0.f, false);
  }
}

// ---------------------------------------------------------------------------
// GCN joint mixing + BN + ReLU -> padded buffer G (B,cout,T+2,J).
// ---------------------------------------------------------------------------
__global__ void gcn_kernel(const float* __restrict__ P,
                           const float* __restrict__ adj,
                           const float* __restrict__ dyn,
                           const float* __restrict__ alpha_p,
                           const float* __restrict__ sb,
                           const float* __restrict__ gg,
                           const float* __restrict__ gb,
                           float* __restrict__ G, int cout) {
  const int b = blockIdx.x, c = blockIdx.y;
  const int tid = threadIdx.x;          // 256
  __shared__ float sA[289], sD[289];
  for (int i = tid; i < 289; i += blockDim.x) {
    sA[i] = adj[i];
    sD[i] = dyn[(size_t)b * 289 + i];
  }
  __syncthreads();
  const float alpha = alpha_p[0];
  const float bias = sb[c], g = gg[c], be = gb[c];
  const float* Pr = P + ((size_t)b * cout + c) * NP;
  float* Gr = G + ((size_t)b * cout + c) * PSTR;

  for (int t = tid; t < TT; t += blockDim.x) {
    float pr[JJ], sf[JJ];
    #pragma unroll
    for (int j = 0; j < JJ; ++j) pr[j] = Pr[t * JJ + j];
    #pragma unroll
    for (int k = 0; k < JJ; ++k) {
      float s = bias;
      #pragma unroll
      for (int j = 0; j < JJ; ++j) s += pr[j] * sA[j * JJ + k];
      sf[k] = s;
    }
    #pragma unroll
    for (int k = 0; k < JJ; ++k) {
      float d = 0.f;
      #pragma unroll
      for (int j = 0; j < JJ; ++j) d += sf[j] * sD[j * JJ + k];
      const float v = g * (sf[k] + alpha * d) + be;
      Gr[(t + 1) * JJ + k] = v > 0.f ? v : 0.f;
    }
  }
  if (tid == 0) {
    #pragma unroll
    for (int k = 0; k < JJ; ++k) Gr[k] = 0.f;
  }
  if (tid == 1) {
    #pragma unroll
    for (int k = 0; k < JJ; ++k) Gr[(TT + 1) * JJ + k] = 0.f;
  }
}

// ---------------------------------------------------------------------------
// Fused temporal conv (3x1, padded) + residual 1x1 conv.  Weights prepacked
// and BN-prescaled: TWP[dt][o][c] = tg[o]*tw[o][c][dt], RWP[o][c]=rg[o]*rw.
// Epilogue: relu(acc + cst[o]) with cst = tg*tb + tbb + rg*rb + rbb.
// ---------------------------------------------------------------------------
template <int CINR, int COUT>
__global__ __launch_bounds__(128) void tcn_kernel(
    const float* __restrict__ G,   const float* __restrict__ TWP,
    const float* __restrict__ Hin, const float* __restrict__ RWP,
    const float* __restrict__ tb,  const float* __restrict__ tg,
    const float* __restrict__ tbb,
    const float* __restrict__ rb,  const float* __restrict__ rg,
    const float* __restrict__ rbb,
    float* __restrict__ Hout) {
  const int lane = threadIdx.x & 31;
  const int wv   = threadIdx.x >> 5;
  const int lo = lane & 15, hi = lane >> 4;
  const int m0 = (blockIdx.x * 4 + wv) * 32;
  const int o0 = blockIdx.y * 64;
  const int b  = blockIdx.z;

  float cst[4];
  #pragma unroll
  for (int s = 0; s < 4; ++s) {
    const int o = o0 + s * 16 + lo;
    cst[s] = tg[o] * tb[o] + tbb[o] + rg[o] * rb[o] + rbb[o];
  }

  v8f acc0[4] = {v8f{}, v8f{}, v8f{}, v8f{}};
  v8f acc1[4] = {v8f{}, v8f{}, v8f{}, v8f{}};

  const float* Gb = G + (size_t)b * COUT * PSTR;
  #pragma unroll
  for (int dt = 0; dt < 3; ++dt) {
    const int pb = m0 + dt * JJ + lo;
    const float* TWd = TWP + (size_t)dt * COUT * COUT;
    #pragma unroll 2
    for (int k0 = 0; k0 < COUT; k0 += 4) {
      const int kb = k0 + hi * 2;
      v2f a0, a1;
      a0.x = Gb[(size_t)kb * PSTR + pb];
      a0.y = Gb[(size_t)(kb + 1) * PSTR + pb];
      a1.x = Gb[(size_t)kb * PSTR + pb + 16];
      a1.y = Gb[(size_t)(kb + 1) * PSTR + pb + 16];
      v2f bw[4];
      #pragma unroll
      for (int s = 0; s < 4; ++s)
        bw[s] = load2(TWd + (o0 + s * 16 + lo) * COUT + kb);
      #pragma unroll
      for (int s = 0; s < 4; ++s) acc0[s] = wmma4(a0, bw[s], acc0[s]);
      #pragma unroll
      for (int s = 0; s < 4; ++s) acc1[s] = wmma4(a1, bw[s], acc1[s]);
    }
  }

  const float* Hb = Hin + (size_t)b * CINR * NP;
  #pragma unroll 2
  for (int k0 = 0; k0 < CINR; k0 += 4) {
    const int kb = k0 + hi * 2;
    v2f a0, a1;
    a0.x = Hb[(size_t)kb * NP + m0 + lo];
    a0.y = Hb[(size_t)(kb + 1) * NP + m0 + lo];
    a1.x = Hb[(size_t)kb * NP + m0 + 16 + lo];
    a1.y = Hb[(size_t)(kb + 1) * NP + m0 + 16 + lo];
    v2f bw[4];
    #pragma unroll
    for (int s = 0; s < 4; ++s)
      bw[s] = load2(RWP + (o0 + s * 16 + lo) * CINR + kb);
    #pragma unroll
    for (int s = 0; s < 4; ++s) acc0[s] = wmma4(a0, bw[s], acc0[s]);
    #pragma unroll
    for (int s = 0; s < 4; ++s) acc1[s] = wmma4(a1, bw[s], acc1[s]);
  }

  #pragma unroll
  for (int s = 0; s < 4; ++s) {
    const int o = o0 + s * 16 + lo;
    float* dst = Hout + ((size_t)b * COUT + o) * NP + m0 + hi * 8;
    store8(dst, acc0[s], cst[s], true);
    store8(dst + 16, acc1[s], cst[s], true);
  }
}

// ---------------------------------------------------------------------------
// Joint pool: pooled[b,c,t] = mean_j h2[b,c,t,j].  LDS-staged for coalescing.
// ---------------------------------------------------------------------------
__global__ void pool_kernel(const float* __restrict__ H2, float* __restrict__ pooled) {
  const int bc  = blockIdx.x;           // b*128 + c
  const int tid = threadIdx.x;          // 256
  __shared__ float sh[NP];
  const float* base = H2 + (size_t)bc * NP;
  for (int i = tid; i < NP; i += blockDim.x) sh[i] = base[i];
  __syncthreads();
  for (int t = tid; t < TT; t += blockDim.x) {
    float s = 0.f;
    #pragma unroll
    for (int j = 0; j < JJ; ++j) s += sh[t * JJ + j];
    pooled[(size_t)bc * TT + t] = s * (1.0f / (float)JJ);
  }
}

// ---------------------------------------------------------------------------
// Projection: out[b,t,e] = sum_c pw[e,c]*pooled[b,c,t] + pb[e]
// Orientation M = e so lanes store 32B contiguous along the embed axis.
// ---------------------------------------------------------------------------
__global__ __launch_bounds__(128) void proj_kernel(const float* __restrict__ pooled,
                                                   const float* __restrict__ PW,
                                                   const float* __restrict__ PB,
                                                   float* __restrict__ out) {
  const int lane = threadIdx.x & 31;
  const int wv   = threadIdx.x >> 5;
  const int lo = lane & 15, hi = lane >> 4;
  const int t0 = (blockIdx.x * 4 + wv) * 64;   // 4 time subtiles per wave
  const int e0 = blockIdx.y * 16;              // embed tile
  const int b  = blockIdx.z;
  constexpr int C = 128, E = 128;

  v8f acc[4] = {v8f{}, v8f{}, v8f{}, v8f{}};
  #pragma unroll 2
  for (int k0 = 0; k0 < C; k0 += 4) {
    const int kb = k0 + hi * 2;
    const v2f a = load2(PW + (e0 + lo) * C + kb);  // contiguous weight pair
    #pragma unroll
    for (int s = 0; s < 4; ++s) {
      const int n = t0 + s * 16 + lo;
      v2f bf;
      bf.x = pooled[((size_t)b * C + kb) * TT + n];
      bf.y = pooled[((size_t)b * C + kb + 1) * TT + n];
      acc[s] = wmma4(a, bf, acc[s]);
    }
  }
  const float bias = PB[e0 + hi * 8 + 0];  // per-r bias differs; handle below
  (void)bias;
  #pragma unroll
  for (int s = 0; s < 4; ++s) {
    const int n = t0 + s * 16 + lo;
    float* dst = out + ((size_t)b * TT + n) * E + e0 + hi * 8;
    float4 v0, v1;
    v0.x = acc[s][0] + PB[e0 + hi * 8 + 0];
    v0.y = acc[s][1] + PB[e0 + hi * 8 + 1];
    v0.z = acc[s][2] + PB[e0 + hi * 8 + 2];
    v0.w = acc[s][3] + PB[e0 + hi * 8 + 3];
    v1.x = acc[s][4] + PB[e0 + hi * 8 + 4];
    v1.y = acc[s][5] + PB[e0 + hi * 8 + 5];
    v1.z = acc[s][6] + PB[e0 + hi * 8 + 6];
    v1.w = acc[s][7] + PB[e0 + hi * 8 + 7];
    *reinterpret_cast<float4*>(dst)     = v0;
    *reinterpret_cast<float4*>(dst + 4) = v1;
  }
}

// ---------------------------------------------------------------------------
// Launcher
// ---------------------------------------------------------------------------
struct BlkP {
  const float *kw, *kb, *qw, *qb, *sw, *sb, *gg, *gb, *al;
  const float *tw, *tb, *tg, *tbb, *rw, *rb, *rg, *rbb;
};

static BlkP get_block(void* const* d_in, int base) {
  BlkP p;
  p.kw  = (const float*)d_in[base + 0];
  p.kb  = (const float*)d_in[base + 1];
  p.qw  = (const float*)d_in[base + 2];
  p.qb  = (const float*)d_in[base + 3];
  p.sw  = (const float*)d_in[base + 4];
  p.sb  = (const float*)d_in[base + 5];
  p.gg  = (const float*)d_in[base + 6];
  p.gb  = (const float*)d_in[base + 7];
  p.al  = (const float*)d_in[base + 8];
  p.tw  = (const float*)d_in[base + 9];
  p.tb  = (const float*)d_in[base + 10];
  p.tg  = (const float*)d_in[base + 11];
  p.tbb = (const float*)d_in[base + 12];
  p.rw  = (const float*)d_in[base + 13];
  p.rb  = (const float*)d_in[base + 14];
  p.rg  = (const float*)d_in[base + 15];
  p.rbb = (const float*)d_in[base + 16];
  return p;
}

extern "C" void kernel_launch(void* const* d_in, const int* in_sizes, int n_in,
                              void* d_out, int out_size, void* d_ws, size_t ws_size,
                              hipStream_t stream) {
  (void)in_sizes; (void)n_in; (void)out_size; (void)ws_size;
  const float* x   = (const float*)d_in[0];
  const float* adj = (const float*)d_in[1];
  const BlkP p0 = get_block(d_in, 2);
  const BlkP p1 = get_block(d_in, 19);
  const float* proj_w = (const float*)d_in[36];
  const float* proj_b = (const float*)d_in[37];
  float* out = (float*)d_out;

  // workspace partition (floats)
  float* ws    = (float*)d_ws;
  float* bufB  = ws;                              // 64*128*NP   (P / h2)
  float* bufD  = bufB + (size_t)64 * 128 * NP;    // 64*128*PSTR (padded gcn)
  float* bufA  = bufD + (size_t)64 * 128 * PSTR;  // 64*64*NP    (h1 / pooled)
  float* xt    = bufA + (size_t)64 * 64 * NP;
  float* kbuf  = xt   + (size_t)64 * 128 * JJ;
  float* qbuf  = kbuf + (size_t)64 * 128 * JJ;
  float* dyn   = qbuf + (size_t)64 * 128 * JJ;
  float* xpad  = dyn  + (size_t)64 * JJ * JJ;     // 64*4*NP
  float* swp0  = xpad + (size_t)64 * 4 * NP;      // 64*4
  float* rwp0  = swp0 + 64 * 4;                   // 64*4   (rg-scaled)
  float* twp0  = rwp0 + 64 * 4;                   // 3*64*64 (tg-scaled)
  float* rwp1  = twp0 + 3 * 64 * 64;              // 128*64  (rg-scaled)
  float* twp1  = rwp1 + 128 * 64;                 // 3*128*128 (tg-scaled)

  // ---- packing (independent, tiny except pack_x) -------------------------
  pack_x_kernel<<<(BB * 4 * NP + 255) / 256, 256, 0, stream>>>(x, xpad);
  pack_w3_kernel<<<1, 256, 0, stream>>>(p0.sw, swp0, nullptr, 0, 64);
  pack_w3_kernel<<<1, 256, 0, stream>>>(p0.rw, rwp0, p0.rg, 1, 64);
  pack_tw_kernel<<<(3 * 64 * 64 + 255) / 256, 256, 0, stream>>>(p0.tw, p0.tg,
                                                                twp0, 64);
  pack_rw_kernel<<<(128 * 64 + 255) / 256, 256, 0, stream>>>(p1.rw, p1.rg,
                                                             rwp1, 64, 128);
  pack_tw_kernel<<<(3 * 128 * 128 + 255) / 256, 256, 0, stream>>>(p1.tw, p1.tg,
                                                                  twp1, 128);

  // ---------------- block 0: cin=3(->4) -> cout=64 ------------------------
  {
    const int cin = 3, cout = 64;
    meanT_kernel<<<BB * cin, 272, 0, stream>>>(x, xt);
    const int tot = BB * cout * JJ;
    kq_kernel<<<(tot + 255) / 256, 256, 0, stream>>>(xt, p0.kw, p0.kb, p0.qw,
                                                     p0.qb, kbuf, qbuf, cin, cout);
    attn_kernel<<<BB, 289, 0, stream>>>(kbuf, qbuf, dyn, cout);
    gemm_p_kernel<4, 64><<<dim3(NP / 128, 1, BB), 128, 0, stream>>>(xpad, swp0,
                                                                    bufB);
    gcn_kernel<<<dim3(BB, cout), 256, 0, stream>>>(bufB, adj, dyn, p0.al, p0.sb,
                                                   p0.gg, p0.gb, bufD, cout);
    tcn_kernel<4, 64><<<dim3(NP / 128, 1, BB), 128, 0, stream>>>(
        bufD, twp0, xpad, rwp0, p0.tb, p0.tg, p0.tbb, p0.rb, p0.rg, p0.rbb,
        bufA);
  }

  // ---------------- block 1: cin=64 -> cout=128 ---------------------------
  {
    const int cin = 64, cout = 128;
    meanT_kernel<<<BB * cin, 272, 0, stream>>>(bufA, xt);
    const int tot = BB * cout * JJ;
    kq_kernel<<<(tot + 255) / 256, 256, 0, stream>>>(xt, p1.kw, p1.kb, p1.qw,
                                                     p1.qb, kbuf, qbuf, cin, cout);
    attn_kernel<<<BB, 289, 0, stream>>>(kbuf, qbuf, dyn, cout);
    gemm_p_kernel<64, 128><<<dim3(NP / 128, 2, BB), 128, 0, stream>>>(bufA, p1.sw,
                                                                      bufB);
    gcn_kernel<<<dim3(BB, cout), 256, 0, stream>>>(bufB, adj, dyn, p1.al, p1.sb,
                                                   p1.gg, p1.gb, bufD, cout);
    tcn_kernel<64, 128><<<dim3(NP / 128, 2, BB), 128, 0, stream>>>(
        bufD, twp1, bufA, rwp1, p1.tb, p1.tg, p1.tbb, p1.rb, p1.rg, p1.rbb,
        bufB);
  }

  // ---------------- pool over joints + projection -------------------------
  pool_kernel<<<BB * 128, 256, 0, stream>>>(bufB, bufA);
  proj_kernel<<<dim3(2, 128 / 16, BB), 128, 0, stream>>>(bufA, proj_w, proj_b,
                                                         out);
}